// DepthReciprocalConv2d_67070209295033
// MI455X (gfx1250) — compile-verified
//
#include <hip/hip_runtime.h>

// Problem constants (fixed by the reference)
#define Bsz 4
#define Cch 64
#define Hh  256
#define Ww  256
#define HWn (Hh * Ww)          // 65536
#define CHWn (Cch * HWn)       // 4194304
#define KW9 576                // 64 in-ch * 9 taps
#define LDP 66                 // padded LDS row stride (dwords): 8B-aligned
                               // fragments, bank-staggered rows

typedef float v2f __attribute__((ext_vector_type(2)));
typedef float v8f __attribute__((ext_vector_type(8)));

// ---------------------------------------------------------------------------
// Kernel 1: s[b,y,x] = FOCAL / (downsample * clip(max_c x, 1, 50)) = 25/depth
// ---------------------------------------------------------------------------
__global__ __launch_bounds__(256) void drc_shift_kernel(
    const float* __restrict__ x, float* __restrict__ s)
{
    int pid = blockIdx.x * 256 + threadIdx.x;        // < B*H*W
    int b   = pid >> 16;                             // / HW
    int rem = pid & (HWn - 1);
    const float* xp = x + (size_t)b * CHWn + rem;
    float m = -3.402823466e+38f;
#pragma unroll 8
    for (int c = 0; c < Cch; ++c)
        m = fmaxf(m, xp[(size_t)c * HWn]);
    float d = fminf(fmaxf(m, 1.0f), 50.0f);
    s[pid] = 25.0f / d;
}

// ---------------------------------------------------------------------------
// Kernel 2: fused dual conv (upper = plain 3x3, lower = warped 3x3) + min.
// One workgroup = 64 pixels of one row (b,y) x all 64 output channels.
// 8 wave32s: wave = (mtile = wid&3 -> 16 outch, half = wid>>2 -> 32 pixels),
// each wave owns 2 N-subtiles x 2 branches of 16x16 f32 accumulators.
// LDS tiles are K-contiguous so each WMMA fragment is one ds_load_b64.
// ---------------------------------------------------------------------------
__global__ __launch_bounds__(256) void drc_conv_kernel(
    const float* __restrict__ x, const float* __restrict__ Wg,
    const float* __restrict__ sbuf, float* __restrict__ out)
{
    __shared__ float sWt[64 * LDP];  // per-tap weights, [co][ci]
    __shared__ float sU [64 * LDP];  // upper im2col patch, [pix][ci]
    __shared__ float sL [64 * LDP];  // lower (warped) patch, [pix][ci]
    __shared__ int   rows4[64][4];   // per-pixel source rows for warp gather
    __shared__ float wts4 [64][4];   // per-pixel gather weights

    const int tid   = threadIdx.x;
    const int xt    = (blockIdx.x & 3) * 64;          // pixel tile base (x)
    const int y     = (blockIdx.x >> 2) & (Hh - 1);
    const int b     =  blockIdx.x >> 10;

    const int lane  = tid & 31;
    const int wid   = tid >> 5;
    const int mtile = wid & 3;           // out-channel tile (16 ch)
    const int half  = wid >> 2;          // which 32-pixel half
    const int l16   = lane & 15;
    const int lh    = lane >> 4;         // half-wave: K pair select
    const int cobase = mtile * 16;

    const float* xb = x + (size_t)b * CHWn;
    const float* sb = sbuf + (size_t)b * HWn;

    v8f accU[2] = {};                    // D = A*B + C accumulators
    v8f accL[2] = {};

    for (int t = 0; t < 9; ++t) {
        const int ky = t / 3, kx = t - 3 * ky;
        const int dy = ky - 1, dx = kx - 1;

        __syncthreads();   // previous tap's LDS reads complete

        // ---- stage per-tap weights K-contiguous: sWt[co][ci] = W[co,ci,ky,kx]
        // (ci fastest across threads -> conflict-free LDS writes)
#pragma unroll 4
        for (int i = 0; i < 16; ++i) {
            int idx = tid + i * 256;
            int ci = idx & 63, co = idx >> 6;
            sWt[co * LDP + ci] = Wg[co * KW9 + ci * 9 + t];
        }

        // ---- per-pixel warp geometry (shared across channels):
        // lower tap = sum_{j<4} wts4[j] * x[c, rows4[j], x+dx]
        if (tid < 64) {
            int p  = tid;
            int xg = xt + p;
            int xx = xg + dx;
            bool vx = (xx >= 0) & (xx < Ww);
            int xxc = min(max(xx, 0), Ww - 1);
            float sc  = sb[y * Ww + xg];          // s at center pixel
            float ysf = (float)y - sc;
            float y0f = floorf(ysf);
            float w   = ysf - y0f;                // shared fractional part
            int  y0b  = (int)y0f;
            float wb0 = 1.0f - w, wb1 = w;
#pragma unroll
            for (int i = 0; i < 2; ++i) {
                float wbase = i ? wb1 : wb0;
                int r = y0b + dy + i;             // x2 source row
                int ra = 0, rb = 0; float wA = 0.f, wB = 0.f;
                if (vx && r >= 0 && r < Hh) {
                    float sr  = sb[r * Ww + xxc]; // s at (r, xx)
                    float yrf = (float)r - sr;
                    float qf  = floorf(yrf);
                    float wq  = yrf - qf;
                    int q = (int)qf;
                    if (q >= 0 && q < Hh)         { ra = q;     wA = wbase * (1.0f - wq); }
                    if (q + 1 >= 0 && q + 1 < Hh) { rb = q + 1; wB = wbase * wq; }
                }
                rows4[p][2 * i]     = ra;  rows4[p][2 * i + 1] = rb;
                wts4 [p][2 * i]     = wA;  wts4 [p][2 * i + 1] = wB;
            }
        }
        __syncthreads();

        // ---- stage im2col patches (upper: shifted read; lower: 4-row gather)
        // p fastest across threads -> coalesced global reads
        const int ry = y + dy;
        const bool vy = (ry >= 0) & (ry < Hh);
#pragma unroll 4
        for (int i = 0; i < 16; ++i) {
            int idx = tid + i * 256;
            int p = idx & 63, c = idx >> 6;
            int xg = xt + p;
            int xx = xg + dx;
            bool vx = (xx >= 0) & (xx < Ww);
            int xxc = min(max(xx, 0), Ww - 1);
            const float* xc = xb + (size_t)c * HWn;

            sU[p * LDP + c] = (vx && vy) ? xc[ry * Ww + xx] : 0.0f;

            float l = wts4[p][0] * xc[rows4[p][0] * Ww + xxc]
                    + wts4[p][1] * xc[rows4[p][1] * Ww + xxc]
                    + wts4[p][2] * xc[rows4[p][2] * Ww + xxc]
                    + wts4[p][3] * xc[rows4[p][3] * Ww + xxc];
            sL[p * LDP + c] = l;
        }
        __syncthreads();

        // ---- WMMA: K-loop over 64 input channels in steps of 4.
        // Fragment layout (ISA): lanes 0-15 carry K={k,k+1}, lanes 16-31
        // K={k+2,k+3}; M/N = lane%16. K-contiguous LDS rows make each
        // fragment a single aligned 8-byte load.
#pragma unroll 4
        for (int k0 = 0; k0 < 64; k0 += 4) {
            int kr = k0 + 2 * lh;
            v2f a = *(const v2f*)(sWt + (cobase + l16) * LDP + kr);
#pragma unroll
            for (int n = 0; n < 2; ++n) {
                int pb = half * 32 + n * 16 + l16;
                v2f bu = *(const v2f*)(sU + pb * LDP + kr);
                v2f bl = *(const v2f*)(sL + pb * LDP + kr);
                accU[n] = __builtin_amdgcn_wmma_f32_16x16x4_f32(
                    false, a, false, bu, (short)0, accU[n], false, false);
                accL[n] = __builtin_amdgcn_wmma_f32_16x16x4_f32(
                    false, a, false, bl, (short)0, accL[n], false, false);
            }
        }
    }

    // ---- epilogue: min(upper, lower); C/D layout: VGPR j -> M=j (+8 for
    // lanes>=16), N = lane%16.
#pragma unroll
    for (int n = 0; n < 2; ++n) {
        int px = xt + half * 32 + n * 16 + l16;
#pragma unroll
        for (int j = 0; j < 8; ++j) {
            int co = cobase + 8 * lh + j;
            out[(size_t)b * CHWn + (size_t)co * HWn + y * Ww + px] =
                fminf(accU[n][j], accL[n][j]);
        }
    }
}

// ---------------------------------------------------------------------------
extern "C" void kernel_launch(void* const* d_in, const int* in_sizes, int n_in,
                              void* d_out, int out_size, void* d_ws, size_t ws_size,
                              hipStream_t stream)
{
    const float* x  = (const float*)d_in[0];   // [4,64,256,256]
    const float* Wg = (const float*)d_in[1];   // [64,64,3,3]
    float* out  = (float*)d_out;               // [4,64,256,256]
    float* sbuf = (float*)d_ws;                // B*H*W floats = 1 MB scratch

    // Kernel 1: per-pixel shift s = 25/clip(max_c x,1,50)
    drc_shift_kernel<<<(Bsz * HWn) / 256, 256, 0, stream>>>(x, sbuf);

    // Kernel 2: fused dual conv + min, WMMA f32 path
    drc_conv_kernel<<<Bsz * Hh * (Ww / 64), 256, 0, stream>>>(x, Wg, sbuf, out);
}